// BatchATSSAssigner_63136019251541
// MI455X (gfx1250) — compile-verified
//
#include <hip/hip_runtime.h>
#include <hip/hip_bf16.h>
#include <float.h>

// ---------------------------------------------------------------------------
// BatchATSSAssigner for MI455X (gfx1250).
// Outputs (concat, float): labels (bs*P) | bboxes (bs*P*4) | scores (bs*P*80) | fg (bs*P)
// ---------------------------------------------------------------------------

#define NUMC 80
#define TOPK 9
#define TK 12              // oversampled top-k for exact refinement
#define MAXCAND 96         // up to ~10 levels * 9

typedef float v2f __attribute__((ext_vector_type(2)));
typedef float v8f __attribute__((ext_vector_type(8)));

// IoU, mmdet BboxOverlaps2D variant: overlap / max(union, 1e-6)
__device__ __forceinline__ float iou_clampeps(float g0, float g1, float g2, float g3,
                                              float p0, float p1, float p2, float p3) {
    float ltx = fmaxf(g0, p0), lty = fmaxf(g1, p1);
    float rbx = fminf(g2, p2), rby = fminf(g3, p3);
    float w = fmaxf(__fsub_rn(rbx, ltx), 0.f);
    float h = fmaxf(__fsub_rn(rby, lty), 0.f);
    float ov = __fmul_rn(w, h);
    float a1 = __fmul_rn(__fsub_rn(g2, g0), __fsub_rn(g3, g1));
    float a2 = __fmul_rn(__fsub_rn(p2, p0), __fsub_rn(p3, p1));
    float un = __fsub_rn(__fadd_rn(a1, a2), ov);
    return __fdiv_rn(ov, fmaxf(un, 1e-6f));
}

// IoU, yolov6 variant: overlap / (union + 1e-9)
__device__ __forceinline__ float iou_addeps(float g0, float g1, float g2, float g3,
                                            float p0, float p1, float p2, float p3) {
    float ltx = fmaxf(g0, p0), lty = fmaxf(g1, p1);
    float rbx = fminf(g2, p2), rby = fminf(g3, p3);
    float w = fmaxf(__fsub_rn(rbx, ltx), 0.f);
    float h = fmaxf(__fsub_rn(rby, lty), 0.f);
    float ov = __fmul_rn(w, h);
    float a1 = __fmul_rn(__fsub_rn(g2, g0), __fsub_rn(g3, g1));
    float a2 = __fmul_rn(__fsub_rn(p2, p0), __fsub_rn(p3, p1));
    float un = __fsub_rn(__fadd_rn(a1, a2), ov);
    return __fdiv_rn(ov, __fadd_rn(un, 1e-9f));
}

// Exact reference-order squared center distance (no FMA contraction).
__device__ __forceinline__ float exact_dd(float gcx, float gcy,
                                          const float* __restrict__ priors, int p) {
    float dx = __fsub_rn(gcx, priors[p * 4 + 0]);
    float dy = __fsub_rn(gcy, priors[p * 4 + 1]);
    return __fadd_rn(__fmul_rn(dx, dx), __fmul_rn(dy, dy));
}

// ---------------------------------------------------------------------------
// Kernel 0: zero the per-prior 64-bit positive masks.
// ---------------------------------------------------------------------------
__global__ __launch_bounds__(256) void zero_kernel(unsigned* q, size_t n) {
    size_t i = (size_t)blockIdx.x * 256 + threadIdx.x;
    if (i < n) q[i] = 0u;
}

// ---------------------------------------------------------------------------
// Kernel 1: pairwise squared center distance via V_WMMA_F32_16X16X4_F32.
// dist2[g][p] = |gc - pc|^2 as rank-4 GEMM with rows [gx,gy,|g|^2,1],
// cols [-2px,-2py,1,|p|^2], coordinates centered at (320,320).
// grid = (numGt/16, bs), 256 threads = 8 waves; waves stride the N tiles.
// ---------------------------------------------------------------------------
__global__ __launch_bounds__(256) void wmma_dist_kernel(
    const float* __restrict__ gt_bboxes, const float* __restrict__ priors,
    float* __restrict__ dist, int numGt, int numPriors) {
    const int b = blockIdx.y;
    const int mt = blockIdx.x;
    const int wave = threadIdx.x >> 5;
    const int lane = threadIdx.x & 31;
    const int l15 = lane & 15;
    const bool hi = lane >= 16;

    // A operand (16x4 f32): lanes 0-15 hold K={0,1}, lanes 16-31 hold K={2,3}.
    const int g = mt * 16 + l15;
    const float* gb = gt_bboxes + ((size_t)b * numGt + g) * 4;
    float gcx = (gb[0] + gb[2]) * 0.5f - 320.f;
    float gcy = (gb[1] + gb[3]) * 0.5f - 320.f;
    float gn = gcx * gcx + gcy * gcy;
    v2f a;
    a.x = hi ? gn : gcx;
    a.y = hi ? 1.0f : gcy;

    const int ntiles = numPriors >> 4;
    for (int nt = wave; nt < ntiles; nt += 8) {
        int p = nt * 16 + l15;
        float px = priors[p * 4 + 0] - 320.f;
        float py = priors[p * 4 + 1] - 320.f;
        float pn = px * px + py * py;
        // B operand (4x16 f32): N = lane&15; lanes 0-15 K={0,1}, lanes 16-31 K={2,3}.
        v2f bv;
        bv.x = hi ? 1.0f : (-2.f * px);
        bv.y = hi ? pn : (-2.f * py);
        v8f c = {0.f, 0.f, 0.f, 0.f, 0.f, 0.f, 0.f, 0.f};
        c = __builtin_amdgcn_wmma_f32_16x16x4_f32(false, a, false, bv, (short)0, c,
                                                  false, false);
        // D layout: VGPR r -> M = r + 8*(lane>=16), N = lane&15.
        size_t rowBase = ((size_t)b * numGt + mt * 16 + (hi ? 8 : 0)) * numPriors + p;
#pragma unroll
        for (int r = 0; r < 8; ++r)
            dist[rowBase + (size_t)r * numPriors] = c[r];
    }
}

// ---------------------------------------------------------------------------
// Kernel 2: per (b, gt): per-level top-12 (approx WMMA dist2 or exact), exact
// re-rank -> top-9 candidates, IoU threshold (mean + std ddof=1), mark
// positives into the per-prior 64-bit mask.
// grid = (numGt, bs), 256 threads.
// ---------------------------------------------------------------------------
__global__ __launch_bounds__(256) void topk_mark_kernel(
    const float* __restrict__ gt_bboxes, const float* __restrict__ priors,
    const float* __restrict__ padFlag, const int* __restrict__ nlp, int nLevels,
    const float* __restrict__ wsDist, unsigned* __restrict__ mLo,
    unsigned* __restrict__ mHi, int numGt, int numPriors, int useApprox) {
    const int b = blockIdx.y;
    const int g = blockIdx.x;
    const int tid = threadIdx.x;

    if (padFlag[(size_t)b * numGt + g] <= 0.f) return;  // uniform exit, no barriers yet

    __shared__ float sD[256 * TK];
    __shared__ int sI[256 * TK];
    __shared__ int candIdx[MAXCAND];
    __shared__ float candOv[MAXCAND];
    __shared__ int candIn[MAXCAND];
    __shared__ float sThr;

    const float* gb = gt_bboxes + ((size_t)b * numGt + g) * 4;
    const float g0 = gb[0], g1 = gb[1], g2 = gb[2], g3 = gb[3];
    // reference: gt center = (lt + rb) * 0.5, elementwise
    const float gcx = __fmul_rn(__fadd_rn(g0, g2), 0.5f);
    const float gcy = __fmul_rn(__fadd_rn(g1, g3), 0.5f);

    int start = 0;
    int candN = 0;
    for (int l = 0; l < nLevels; ++l) {
        const int cnt = nlp[l];
        const int ktake = cnt < TOPK ? cnt : TOPK;

        // --- local top-TK scan (sorted ascending by (d, idx)) ---
        float td[TK];
        int ti[TK];
#pragma unroll
        for (int k = 0; k < TK; ++k) { td[k] = FLT_MAX; ti[k] = 0x7FFFFFFF; }
        for (int j = tid; j < cnt; j += 256) {
            int p = start + j;
            float d;
            if (useApprox)
                d = wsDist[((size_t)b * numGt + g) * numPriors + p];
            else
                d = exact_dd(gcx, gcy, priors, p);
            if (d < td[TK - 1] || (d == td[TK - 1] && p < ti[TK - 1])) {
                float cd = d; int cp = p;
#pragma unroll
                for (int k = 0; k < TK; ++k) {
                    bool less = (cd < td[k]) || (cd == td[k] && cp < ti[k]);
                    if (less) {
                        float tf = td[k]; int tp = ti[k];
                        td[k] = cd; ti[k] = cp;
                        cd = tf; cp = tp;
                    }
                }
            }
        }

        __syncthreads();  // protect previous level's refine reads of sD/sI
#pragma unroll
        for (int k = 0; k < TK; ++k) { sD[tid * TK + k] = td[k]; sI[tid * TK + k] = ti[k]; }
        __syncthreads();

        // --- tree merge of 256 sorted lists -> thread 0 holds global top-TK ---
        for (int off = 128; off >= 1; off >>= 1) {
            if (tid < off) {
                int A = tid * TK, B = (tid + off) * TK;
                int i = 0, j = 0;
                float od[TK]; int oi[TK];
#pragma unroll
                for (int k = 0; k < TK; ++k) {
                    float da = sD[A + i]; int pa = sI[A + i];
                    float db = sD[B + j]; int pb = sI[B + j];
                    bool takeA = (da < db) || (da == db && pa <= pb);
                    if (takeA) { od[k] = da; oi[k] = pa; ++i; }
                    else       { od[k] = db; oi[k] = pb; ++j; }
                }
#pragma unroll
                for (int k = 0; k < TK; ++k) { sD[A + k] = od[k]; sI[A + k] = oi[k]; }
            }
            __syncthreads();
        }

        // --- thread 0: exact re-rank of the TK survivors, emit top-ktake ---
        if (tid == 0) {
            for (int k = 0; k < TK; ++k) {
                int p = sI[k];
                sD[k] = (p == 0x7FFFFFFF) ? FLT_MAX : exact_dd(gcx, gcy, priors, p);
            }
            for (int r = 0; r < ktake; ++r) {
                float bd = FLT_MAX; int bi = 0x7FFFFFFF; int bm = 0;
                for (int m = 0; m < TK; ++m) {
                    if (sD[m] < bd || (sD[m] == bd && sI[m] < bi)) {
                        bd = sD[m]; bi = sI[m]; bm = m;
                    }
                }
                candIdx[candN + r] = bi;
                sD[bm] = FLT_MAX; sI[bm] = 0x7FFFFFFF;
            }
        }
        candN += ktake;
        start += cnt;
    }
    __syncthreads();

    // --- candidate IoUs (prior cell box vs gt) + in-gt test ---
    if (tid < candN) {
        int p = candIdx[tid];
        float px = priors[p * 4 + 0], py = priors[p * 4 + 1];
        float hw = __fmul_rn(priors[p * 4 + 2], 2.5f);
        float hh = __fmul_rn(priors[p * 4 + 3], 2.5f);
        float p0 = __fsub_rn(px, hw), p1 = __fsub_rn(py, hh);
        float p2 = __fadd_rn(px, hw), p3 = __fadd_rn(py, hh);
        candOv[tid] = iou_clampeps(g0, g1, g2, g3, p0, p1, p2, p3);
        // prior center == priors[:, :2] exactly (grid coords are exact floats)
        float m0 = __fsub_rn(px, g0), m1 = __fsub_rn(py, g1);
        float m2 = __fsub_rn(g2, px), m3 = __fsub_rn(g3, py);
        float mn = fminf(fminf(m0, m1), fminf(m2, m3));
        candIn[tid] = (mn > 1e-9f) ? 1 : 0;
    }
    __syncthreads();

    if (tid == 0) {
        float s = 0.f;
        for (int i = 0; i < candN; ++i) s += candOv[i];
        float mean = s / (float)candN;
        float v = 0.f;
        for (int i = 0; i < candN; ++i) {
            float d = candOv[i] - mean;
            v += d * d;
        }
        sThr = mean + sqrtf(v / (float)(candN - 1));
    }
    __syncthreads();

    if (tid < candN && candOv[tid] > sThr && candIn[tid]) {
        int p = candIdx[tid];
        if (g < 32) atomicOr(&mLo[(size_t)b * numPriors + p], 1u << g);
        else        atomicOr(&mHi[(size_t)b * numPriors + p], 1u << (g - 32));
    }
}

// ---------------------------------------------------------------------------
// Kernel 3: per prior — conflict resolution, gather labels/boxes, pred-IoU
// weight, one-hot score write.  grid = (ceil(P/256), bs), 256 threads.
// ---------------------------------------------------------------------------
__global__ __launch_bounds__(256) void out_kernel(
    const float* __restrict__ pred, const float* __restrict__ priors,
    const int* __restrict__ gtLab, const float* __restrict__ gtBox,
    const unsigned* __restrict__ mLo, const unsigned* __restrict__ mHi,
    float* __restrict__ outL, float* __restrict__ outB, float* __restrict__ outS,
    float* __restrict__ outF, int numGt, int numPriors) {
    const int b = blockIdx.y;
    const int tid = threadIdx.x;
    const int p0 = blockIdx.x * 256;

    __shared__ float sGt[256 * 4];
    __shared__ int sLab[256];
    __shared__ float sW[256];
    __shared__ int sL[256];

    for (int i = tid; i < numGt * 4; i += 256) sGt[i] = gtBox[(size_t)b * numGt * 4 + i];
    for (int i = tid; i < numGt; i += 256) sLab[i] = gtLab[(size_t)b * numGt + i];
    __syncthreads();

    const int p = p0 + tid;
    const bool valid = p < numPriors;
    float w = 0.f;
    int lab = -1;

    if (valid) {
        unsigned lo = mLo[(size_t)b * numPriors + p];
        unsigned hm = mHi[(size_t)b * numPriors + p];
        int fg = __popc(lo) + __popc(hm);
        int gi = 0;
        if (fg == 1) {
            gi = lo ? (__ffs(lo) - 1) : (32 + __ffs(hm) - 1);
        } else if (fg > 1) {
            // argmax over ALL gts of prior-cell IoU (first max, like jnp.argmax)
            float px = priors[p * 4 + 0], py = priors[p * 4 + 1];
            float hw = __fmul_rn(priors[p * 4 + 2], 2.5f);
            float hh = __fmul_rn(priors[p * 4 + 3], 2.5f);
            float q0 = __fsub_rn(px, hw), q1 = __fsub_rn(py, hh);
            float q2 = __fadd_rn(px, hw), q3 = __fadd_rn(py, hh);
            float best = -1.f;
            for (int gg = 0; gg < numGt; ++gg) {
                float v = iou_clampeps(sGt[gg * 4 + 0], sGt[gg * 4 + 1],
                                       sGt[gg * 4 + 2], sGt[gg * 4 + 3], q0, q1, q2, q3);
                if (v > best) { best = v; gi = gg; }
            }
        }
        if (fg > 0) {
            lab = sLab[gi];
            const float* pd = pred + ((size_t)b * numPriors + p) * 4;
            w = iou_addeps(sGt[gi * 4 + 0], sGt[gi * 4 + 1], sGt[gi * 4 + 2],
                           sGt[gi * 4 + 3], pd[0], pd[1], pd[2], pd[3]);
        }
        outL[(size_t)b * numPriors + p] = (fg > 0) ? (float)lab : (float)NUMC;
        outF[(size_t)b * numPriors + p] = (fg > 0) ? 1.f : 0.f;
#pragma unroll
        for (int i = 0; i < 4; ++i)
            outB[((size_t)b * numPriors + p) * 4 + i] = sGt[gi * 4 + i];
    }

    sW[tid] = (valid && lab >= 0) ? w : 0.f;
    sL[tid] = (valid && lab >= 0) ? lab : -1;
    __syncthreads();

    // Coalesced one-hot score write: 256*80 contiguous floats per block.
    for (int idx = tid; idx < 256 * NUMC; idx += 256) {
        int local = idx / NUMC;
        int cls = idx - local * NUMC;
        int pp = p0 + local;
        if (pp < numPriors)
            outS[((size_t)b * numPriors + pp) * NUMC + cls] =
                (cls == sL[local]) ? sW[local] : 0.f;
    }
}

// ---------------------------------------------------------------------------
extern "C" void kernel_launch(void* const* d_in, const int* in_sizes, int n_in,
                              void* d_out, int out_size, void* d_ws, size_t ws_size,
                              hipStream_t stream) {
    const float* pred   = (const float*)d_in[0];
    const float* priors = (const float*)d_in[1];
    const int*   gtLab  = (const int*)d_in[2];
    const float* gtBox  = (const float*)d_in[3];
    const float* padF   = (const float*)d_in[4];
    const int*   nlp    = (const int*)d_in[5];
    const int nLevels   = in_sizes[5];

    const int numPriors = in_sizes[1] / 4;
    const int bs        = in_sizes[0] / (numPriors * 4);
    const int numGt     = in_sizes[3] / (bs * 4);

    const size_t maskElems = (size_t)bs * numPriors;
    unsigned* mLo = (unsigned*)d_ws;
    unsigned* mHi = mLo + maskElems;
    float* wsDist = (float*)(mHi + maskElems);
    const size_t need = maskElems * 2 * sizeof(unsigned) +
                        (size_t)bs * numGt * numPriors * sizeof(float);
    const int useApprox = (ws_size >= need) && (numGt % 16 == 0) &&
                          (numPriors % 16 == 0) && (numGt <= 256);

    // 0) zero masks
    {
        size_t n = maskElems * 2;
        int blocks = (int)((n + 255) / 256);
        zero_kernel<<<blocks, 256, 0, stream>>>(mLo, n);
    }
    // 1) WMMA pairwise squared distances (bulk ranking data)
    if (useApprox) {
        dim3 grid(numGt / 16, bs);
        wmma_dist_kernel<<<grid, 256, 0, stream>>>(gtBox, priors, wsDist, numGt,
                                                   numPriors);
    }
    // 2) per-gt top-k + threshold + positive marking
    {
        dim3 grid(numGt, bs);
        topk_mark_kernel<<<grid, 256, 0, stream>>>(gtBox, priors, padF, nlp, nLevels,
                                                   wsDist, mLo, mHi, numGt, numPriors,
                                                   useApprox);
    }
    // 3) per-prior outputs
    {
        float* outL = (float*)d_out;
        float* outB = outL + (size_t)bs * numPriors;
        float* outS = outB + (size_t)bs * numPriors * 4;
        float* outF = outS + (size_t)bs * numPriors * NUMC;
        dim3 grid((numPriors + 255) / 256, bs);
        out_kernel<<<grid, 256, 0, stream>>>(pred, priors, gtLab, gtBox, mLo, mHi,
                                             outL, outB, outS, outF, numGt, numPriors);
    }
}